// BackboneTrajectoryLoss_34540126994455
// MI455X (gfx1250) — compile-verified
//
#include <hip/hip_runtime.h>
#include <math.h>

typedef __attribute__((ext_vector_type(2))) float v2f;
typedef __attribute__((ext_vector_type(8))) float v8f;

#define L_LAYERS 8
#define BATCH    2
#define NRES     1024
#define KPAD     32          // 28 real features padded to 32 (8 x K=4 WMMA steps)
#define FAPE_EPS 1e-4f

// ---------------------------------------------------------------------------
// Stage 1: build frame features G'[lb][f][32] and atom features X'[lb][a][32]
// err2[f,a] = sum_k FF[f,k] * XF[a,k]
// ---------------------------------------------------------------------------
__global__ void fape_features_kernel(const float* __restrict__ pR,   // [L,B,N,3,3]
                                     const float* __restrict__ pT,   // [L,B,N,3]
                                     const float* __restrict__ tR,   // [B,N,3,3]
                                     const float* __restrict__ tT,   // [B,N,3]
                                     float* __restrict__ FF,         // [16,N,32]
                                     float* __restrict__ XF)         // [16,N,32]
{
    const int n  = blockIdx.x * blockDim.x + threadIdx.x;   // residue 0..1023
    const int lb = blockIdx.y;                              // 0..15
    const int b  = lb & (BATCH - 1);
    const long pi = (long)lb * NRES + n;                    // pred index
    const long ti = (long)b  * NRES + n;                    // true index

    float Rp[3][3], Rt[3][3], tp[3], tt[3];
#pragma unroll
    for (int i = 0; i < 3; ++i) {
#pragma unroll
        for (int j = 0; j < 3; ++j) {
            Rp[i][j] = pR[pi * 9 + i * 3 + j];
            Rt[i][j] = tR[ti * 9 + i * 3 + j];
        }
        tp[i] = pT[pi * 3 + i];
        tt[i] = tT[ti * 3 + i];
    }

    // M (3x6) = [Rp^T | -Rt^T];  e = (-Rp^T tp) - (-Rt^T tt)
    float M[3][6], e[3];
#pragma unroll
    for (int r = 0; r < 3; ++r) {
#pragma unroll
        for (int i = 0; i < 3; ++i) {
            M[r][i]     =  Rp[i][r];
            M[r][i + 3] = -Rt[i][r];
        }
        float c = 0.f, d = 0.f;
#pragma unroll
        for (int j = 0; j < 3; ++j) {
            c -= Rp[j][r] * tp[j];
            d -= Rt[j][r] * tt[j];
        }
        e[r] = c - d;
    }

    float ff[KPAD];
    float xf[KPAD];
    float x[6] = { tp[0], tp[1], tp[2], tt[0], tt[1], tt[2] };

    int k = 0;
#pragma unroll
    for (int i = 0; i < 6; ++i) {
#pragma unroll
        for (int j = i; j < 6; ++j) {
            float g = M[0][i] * M[0][j] + M[1][i] * M[1][j] + M[2][i] * M[2][j];
            ff[k] = (i == j) ? g : 2.0f * g;
            xf[k] = x[i] * x[j];
            ++k;
        }
    }
#pragma unroll
    for (int i = 0; i < 6; ++i) {
        ff[k] = 2.0f * (M[0][i] * e[0] + M[1][i] * e[1] + M[2][i] * e[2]);
        xf[k] = x[i];
        ++k;
    }
    ff[k] = e[0] * e[0] + e[1] * e[1] + e[2] * e[2];
    xf[k] = 1.0f;
    ++k;
#pragma unroll
    for (; k < KPAD; ++k) { ff[k] = 0.0f; xf[k] = 0.0f; }

    float* ffo = FF + ((long)lb * NRES + n) * KPAD;
    float* xfo = XF + ((long)lb * NRES + n) * KPAD;
#pragma unroll
    for (int i = 0; i < KPAD; ++i) { ffo[i] = ff[i]; xfo[i] = xf[i]; }
}

// ---------------------------------------------------------------------------
// Stage 2: per (lb, frame-tile) wave: GEMM err2 = FF * XF^T via
// V_WMMA_F32_16X16X4_F32 chain (K=32), fused fast-sqrt/clamp/mask epilogue,
// wave reduction, atomicAdd into partial[lb].
//
// A-matrix 16x4 f32 layout: lanes 0-15 hold K={4i,4i+1}, lanes 16-31 hold
// K={4i+2,4i+3} of row (lane&15). B (4x16) symmetric with N=(lane&15).
// C/D: lane -> N=(lane&15), VGPR r -> M = r + 8*(lane>=16).
// ---------------------------------------------------------------------------
__global__ void __launch_bounds__(32)
fape_pair_wmma_kernel(const float* __restrict__ FF,     // [16,N,32]
                      const float* __restrict__ XF,     // [16,N,32]
                      const float* __restrict__ mask,   // [B,N]
                      float* __restrict__ partial)      // [16]
{
    const int lane  = threadIdx.x;            // 0..31
    const int ftile = blockIdx.x;             // 0..63
    const int lb    = blockIdx.y;             // 0..15
    const int b     = lb & (BATCH - 1);

    const float* FF_lb = FF + (long)lb * NRES * KPAD;
    const float* XF_lb = XF + (long)lb * NRES * KPAD;

    const int hi   = lane >> 4;               // lane group
    const int kgrp = hi * 2;                  // K sub-offset within a K=4 step
    const int frow = ftile * 16 + (lane & 15);

    // A tile: frame features for 16 rows, loaded once.
    v2f a[8];
#pragma unroll
    for (int i = 0; i < 8; ++i)
        a[i] = *(const v2f*)(FF_lb + (long)frow * KPAD + (i << 2) + kgrp);

    // Frame masks for this lane's C rows: M = r + 8*hi
    const int fbase = ftile * 16 + (hi ? 8 : 0);
    float mF[8];
#pragma unroll
    for (int r = 0; r < 8; ++r)
        mF[r] = mask[b * NRES + fbase + r];

    float acc = 0.0f;

    for (int atile = 0; atile < NRES / 16; ++atile) {
        const int acol = atile * 16 + (lane & 15);

        v2f bb[8];
#pragma unroll
        for (int i = 0; i < 8; ++i)
            bb[i] = *(const v2f*)(XF_lb + (long)acol * KPAD + (i << 2) + kgrp);

        v8f c = {};
#pragma unroll
        for (int i = 0; i < 8; ++i)
            c = __builtin_amdgcn_wmma_f32_16x16x4_f32(
                    /*neg_a=*/false, a[i], /*neg_b=*/false, bb[i],
                    /*c_mod=*/(short)0, c, /*reuse_a=*/false, /*reuse_b=*/false);

        const float mA = mask[b * NRES + acol];
#pragma unroll
        for (int r = 0; r < 8; ++r) {
            float v = fmaxf(c[r], 0.0f) + FAPE_EPS;  // guard against fp cancellation
            // Hardware v_sqrt_f32 (TRANS op, co-executes with WMMA); ~1 ulp is
            // far below the clamp/scale resolution of this loss.
            float e = __builtin_amdgcn_sqrtf(v);
            e = fminf(e, 10.0f) * 0.1f;              // clamp to D_CLAMP, divide by Z
            acc += e * mF[r] * mA;
        }
    }

    // wave32 reduction
#pragma unroll
    for (int off = 16; off >= 1; off >>= 1)
        acc += __shfl_xor(acc, off, 32);

    if (lane == 0)
        atomicAdd(&partial[lb], acc);
}

// ---------------------------------------------------------------------------
// Stage 3: mask sums per batch, final normalization + layer mean.
// ---------------------------------------------------------------------------
__global__ void fape_finalize_kernel(const float* __restrict__ mask,     // [B,N]
                                     const float* __restrict__ partial,  // [16]
                                     float* __restrict__ out)            // [B]
{
    __shared__ float red[256];
    const int t = threadIdx.x;        // 256 threads: [0,128)->b=0, [128,256)->b=1
    const int b = t >> 7;
    const int i = t & 127;

    float s = 0.0f;
    for (int k = 0; k < NRES; k += 128)
        s += mask[b * NRES + i + k];
    red[t] = s;
    __syncthreads();

    for (int off = 64; off >= 1; off >>= 1) {
        if (i < off) red[t] += red[t + off];
        __syncthreads();
    }

    if (t == 0) {
        for (int bb = 0; bb < BATCH; ++bb) {
            const float denom = FAPE_EPS + red[bb * 128];
            float accum = 0.0f;
            for (int l = 0; l < L_LAYERS; ++l)
                accum += partial[l * BATCH + bb];
            out[bb] = accum / (denom * denom) / (float)L_LAYERS;
        }
    }
}

// ---------------------------------------------------------------------------
extern "C" void kernel_launch(void* const* d_in, const int* in_sizes, int n_in,
                              void* d_out, int out_size, void* d_ws, size_t ws_size,
                              hipStream_t stream)
{
    const float* pR   = (const float*)d_in[0];  // traj_rotations    [L,B,N,3,3]
    const float* pT   = (const float*)d_in[1];  // traj_translations [L,B,N,3]
    const float* tR   = (const float*)d_in[2];  // true_rotations    [B,N,3,3]
    const float* tT   = (const float*)d_in[3];  // true_translations [B,N,3]
    const float* mask = (const float*)d_in[4];  // backbone_mask     [B,N]
    float* out = (float*)d_out;

    const long featElems = (long)L_LAYERS * BATCH * NRES * KPAD;  // 524288
    float* FF      = (float*)d_ws;
    float* XF      = FF + featElems;
    float* partial = XF + featElems;

    hipMemsetAsync(partial, 0, L_LAYERS * BATCH * sizeof(float), stream);

    fape_features_kernel<<<dim3(NRES / 256, L_LAYERS * BATCH), 256, 0, stream>>>(
        pR, pT, tR, tT, FF, XF);

    fape_pair_wmma_kernel<<<dim3(NRES / 16, L_LAYERS * BATCH), 32, 0, stream>>>(
        FF, XF, mask, partial);

    fape_finalize_kernel<<<1, 256, 0, stream>>>(mask, partial, out);
}